// LinearAttention_78391743087168
// MI455X (gfx1250) — compile-verified
//
#include <hip/hip_runtime.h>

typedef __attribute__((ext_vector_type(2))) float v2f;
typedef __attribute__((ext_vector_type(4))) float v4f;
typedef __attribute__((ext_vector_type(8))) float v8f;

#define BATCH 4
#define NSEQ  16384
#define CDIM  1024
#define HEADS 8
#define DH    64        // DIM_HEAD
#define QKV3  192       // 3*DH
#define SCALE 0.125f    // DIM_HEAD ** -0.5

// Async global->LDS 16B transfer: IOFFSET applies to BOTH lds and global addr
// (ISA 15.18.3), so one addr setup serves a contiguous run of 16B moves.
#define ASYNC_B128(lds, g, off)                                            \
    asm volatile("global_load_async_to_lds_b128 %0, %1, off offset:" #off  \
                 :: "v"(lds), "v"(g) : "memory")
#define WAIT_ASYNC0() asm volatile("s_wait_asynccnt 0x0" ::: "memory")

// ---------------------------------------------------------------------------
// Kernel 0: zero the context accumulator
// ---------------------------------------------------------------------------
__global__ void k_zero(float* __restrict__ p, int n) {
    int i = blockIdx.x * blockDim.x + threadIdx.x;
    if (i < n) p[i] = 0.0f;
}

// ---------------------------------------------------------------------------
// Kernel 1: qkv = x @ Wqkv (WMMA f32 16x16x4) on a 64-row x 192-col tile with
// double-buffered async global->LDS staging of the x K-chunks; then per-head
// softmax of k (d=8), context partials -> atomicAdd, scaled q -> qws.
// Block: 128 threads (4 waves). Wave w: col tiles 3w..3w+2, all 4 row tiles.
// ---------------------------------------------------------------------------
__global__ __launch_bounds__(128) void k_qkv_ctx(
    const float* __restrict__ x, const float* __restrict__ Wqkv,
    float* __restrict__ qws, float* __restrict__ ctx)
{
    __shared__ float xs[2][64][68];  // double-buffered 64x64 K-chunk (padded)
    __shared__ float qkvs[64][196];  // 64 x 192 output tile (padded stride)

    const int tid  = threadIdx.x;
    const int lane = tid & 31;
    const int wave = tid >> 5;       // 0..3
    const int half = lane >> 4;      // selects K pair / row half of fragments
    const int l16  = lane & 15;
    const long rowbase = (long)blockIdx.x * 64;

    // this thread's staging slot: 32 consecutive floats of one row
    const int sr  = tid >> 1;          // 0..63
    const int sc0 = (tid & 1) * 32;    // 0 or 32

    // issue async copy of x[rowbase+sr][kk+sc0 .. +31] into xs[bufi]
    auto issue_chunk = [&](int kk, int bufi) {
        const unsigned long long g =
            (unsigned long long)(size_t)(x + (rowbase + sr) * (long)CDIM + kk + sc0);
        const unsigned lds = (unsigned)(size_t)&xs[bufi][sr][sc0];
        ASYNC_B128(lds, g, 0);
        ASYNC_B128(lds, g, 16);
        ASYNC_B128(lds, g, 32);
        ASYNC_B128(lds, g, 48);
        ASYNC_B128(lds, g, 64);
        ASYNC_B128(lds, g, 80);
        ASYNC_B128(lds, g, 96);
        ASYNC_B128(lds, g, 112);
    };

    v8f acc[4][3] = {};

    issue_chunk(0, 0);
    int buf = 0;
    for (int kk = 0; kk < CDIM; kk += 64) {
        WAIT_ASYNC0();
        __syncthreads();
        if (kk + 64 < CDIM) issue_chunk(kk + 64, buf ^ 1);   // prefetch next
        #pragma unroll
        for (int k = 0; k < 64; k += 4) {
            // A fragments for the 4 row tiles (shared across 3 col tiles)
            v2f A[4];
            #pragma unroll
            for (int rt = 0; rt < 4; rt++)
                A[rt] = *(const v2f*)&xs[buf][16 * rt + l16][k + 2 * half];
            #pragma unroll
            for (int t = 0; t < 3; t++) {
                const int col = (wave * 3 + t) * 16 + l16;
                const float* wp = Wqkv + (long)(kk + k + 2 * half) * QKV3 + col;
                v2f Bv;
                Bv.x = wp[0];
                Bv.y = wp[QKV3];
                #pragma unroll
                for (int rt = 0; rt < 4; rt++)
                    acc[rt][t] = __builtin_amdgcn_wmma_f32_16x16x4_f32(
                        false, A[rt], false, Bv, (short)0, acc[rt][t], false, false);
            }
        }
        buf ^= 1;
    }
    __syncthreads();

    // spill accumulators: lane holds col l16; VGPR i = row i + 8*half
    #pragma unroll
    for (int rt = 0; rt < 4; rt++) {
        #pragma unroll
        for (int t = 0; t < 3; t++) {
            const int col = (wave * 3 + t) * 16 + l16;
            #pragma unroll
            for (int i = 0; i < 8; i++)
                qkvs[16 * rt + i + 8 * half][col] = acc[rt][t][i];
        }
    }
    __syncthreads();

    // softmax of k over per-head dim (8): 64 rows x 8 heads = 512 pairs
    #pragma unroll
    for (int p = 0; p < 4; p++) {
        const int id = tid + 128 * p;
        const int r = id & 63;
        const int h = id >> 6;
        float kv[8];
        float m = -3.4e38f;
        #pragma unroll
        for (int j = 0; j < 8; j++) { kv[j] = qkvs[r][DH + h * 8 + j]; m = fmaxf(m, kv[j]); }
        float s = 0.0f;
        #pragma unroll
        for (int j = 0; j < 8; j++) { kv[j] = __expf(kv[j] - m); s += kv[j]; }
        const float inv = 1.0f / s;
        #pragma unroll
        for (int j = 0; j < 8; j++) qkvs[r][DH + h * 8 + j] = kv[j] * inv;
    }
    __syncthreads();

    // context partials: thread owns (h, dk, row-half), accumulates all 8 dv.
    {
        const int b  = (int)(rowbase / NSEQ);
        const int h  = tid >> 4;               // 0..7
        const int dk = tid & 7;                // 0..7
        const int rh = (tid >> 3) & 1;         // row half: 0 or 1
        v4f s0 = {}, s1 = {};
        for (int r = rh * 32; r < rh * 32 + 32; r++) {
            const float kval = qkvs[r][DH + h * 8 + dk];
            const v4f v0 = *(const v4f*)&qkvs[r][2 * DH + h * 8];
            const v4f v1 = *(const v4f*)&qkvs[r][2 * DH + h * 8 + 4];
            s0 += kval * v0;
            s1 += kval * v1;
        }
        float* cp = &ctx[((b * HEADS + h) * 8 + dk) * 8];
        #pragma unroll
        for (int e = 0; e < 4; e++) atomicAdd(cp + e, s0[e]);
        #pragma unroll
        for (int e = 0; e < 4; e++) atomicAdd(cp + 4 + e, s1[e]);
    }

    // write scaled q tile to workspace (coalesced v4f stores)
    {
        float* dst = qws + (rowbase + sr) * (long)DH + sc0;
        #pragma unroll
        for (int j = 0; j < 8; j++) {
            v4f q = *(const v4f*)&qkvs[sr][sc0 + 4 * j];
            q *= SCALE;
            *(v4f*)(dst + 4 * j) = q;
        }
    }
}

// ---------------------------------------------------------------------------
// Kernel 2: Wfused[b] = blockdiag(ctx_b) @ Wout  -> [B][64][1024]
// ---------------------------------------------------------------------------
__global__ __launch_bounds__(256) void k_fuse(
    const float* __restrict__ ctx, const float* __restrict__ Wout,
    float* __restrict__ Wf)
{
    const int b = blockIdx.x;
    const int t = threadIdx.x;
    __shared__ float cs[512];
    cs[t]       = ctx[b * 512 + t];
    cs[t + 256] = ctx[b * 512 + t + 256];
    __syncthreads();
    for (int c = t; c < CDIM; c += 256) {
        for (int j = 0; j < DH; j++) {
            const int h = j >> 3, dk = j & 7;
            float s = 0.0f;
            #pragma unroll
            for (int e = 0; e < 8; e++)
                s += cs[(h * 8 + dk) * 8 + e] * Wout[(h * 8 + e) * CDIM + c];
            Wf[((long)b * DH + j) * CDIM + c] = s;
        }
    }
}

// ---------------------------------------------------------------------------
// Kernel 3: y = q @ Wfused[b] + bout  (WMMA f32 16x16x4)
// Block: 256 threads (8 waves). Block tile: 64 rows x 512 cols (grid.y = 2).
// Wave w: 4 row tiles x 4 col tiles = 16 accumulators. q tile staged via
// async global->LDS.
// ---------------------------------------------------------------------------
__global__ __launch_bounds__(256) void k_out(
    const float* __restrict__ qws, const float* __restrict__ Wf,
    const float* __restrict__ bout, float* __restrict__ y)
{
    __shared__ float qs[64][68];   // padded

    const int tid  = threadIdx.x;
    const int lane = tid & 31;
    const int wave = tid >> 5;     // 0..7
    const int half = lane >> 4;
    const int l16  = lane & 15;
    const long rowbase = (long)blockIdx.x * 64;
    const int  colhalf = blockIdx.y;             // cols 0-511 / 512-1023
    const int  b = (int)(rowbase / NSEQ);
    const float* Wb = Wf + (long)b * DH * CDIM;

    // async-stage 64x64 q tile: thread moves 16 consecutive floats (64B)
    {
        const int r  = tid >> 2;          // 0..63
        const int c0 = (tid & 3) * 16;    // 0,16,32,48
        const unsigned long long g =
            (unsigned long long)(size_t)(qws + (rowbase + r) * (long)DH + c0);
        const unsigned lds = (unsigned)(size_t)&qs[r][c0];
        ASYNC_B128(lds, g, 0);
        ASYNC_B128(lds, g, 16);
        ASYNC_B128(lds, g, 32);
        ASYNC_B128(lds, g, 48);
    }
    WAIT_ASYNC0();
    __syncthreads();

    v8f acc[4][4] = {};
    #pragma unroll
    for (int k = 0; k < DH; k += 4) {
        v2f A[4];
        #pragma unroll
        for (int rt = 0; rt < 4; rt++)
            A[rt] = *(const v2f*)&qs[16 * rt + l16][k + 2 * half];
        #pragma unroll
        for (int t = 0; t < 4; t++) {
            const int col = colhalf * 512 + wave * 64 + t * 16 + l16;
            const float* wp = Wb + (long)(k + 2 * half) * CDIM + col;
            v2f Bv;
            Bv.x = wp[0];
            Bv.y = wp[CDIM];
            #pragma unroll
            for (int rt = 0; rt < 4; rt++)
                acc[rt][t] = __builtin_amdgcn_wmma_f32_16x16x4_f32(
                    false, A[rt], false, Bv, (short)0, acc[rt][t], false, false);
        }
    }

    // bias + store (each b32 store covers two full 64B lines)
    #pragma unroll
    for (int t = 0; t < 4; t++) {
        const int col = colhalf * 512 + wave * 64 + t * 16 + l16;
        const float bb = bout[col];
        #pragma unroll
        for (int rt = 0; rt < 4; rt++) {
            #pragma unroll
            for (int i = 0; i < 8; i++)
                y[(rowbase + 16 * rt + i + 8 * half) * (long)CDIM + col] = acc[rt][t][i] + bb;
        }
    }
}

// ---------------------------------------------------------------------------
extern "C" void kernel_launch(void* const* d_in, const int* in_sizes, int n_in,
                              void* d_out, int out_size, void* d_ws, size_t ws_size,
                              hipStream_t stream) {
    const float* x    = (const float*)d_in[0];
    const float* Wqkv = (const float*)d_in[1];
    const float* Wout = (const float*)d_in[2];
    const float* bout = (const float*)d_in[3];
    float* y = (float*)d_out;

    // workspace layout
    float* qws = (float*)d_ws;                              // 65536*64 floats (16 MB)
    float* ctx = qws + (size_t)BATCH * NSEQ * DH;           // 2048 floats
    float* Wf  = ctx + (size_t)BATCH * HEADS * 8 * 8;       // 4*64*1024 floats (1 MB)

    k_zero<<<(2048 + 255) / 256, 256, 0, stream>>>(ctx, 2048);
    k_qkv_ctx<<<BATCH * NSEQ / 64, 128, 0, stream>>>(x, Wqkv, qws, ctx);
    k_fuse<<<BATCH, 256, 0, stream>>>(ctx, Wout, Wf);
    k_out<<<dim3(BATCH * NSEQ / 64, 2), 256, 0, stream>>>(qws, Wf, bout, y);
}